// hprompt_gcn_21534966022319
// MI455X (gfx1250) — compile-verified
//
#include <hip/hip_runtime.h>
#include <hip/hip_bf16.h>

typedef __attribute__((ext_vector_type(2))) float v2f;
typedef __attribute__((ext_vector_type(8))) float v8f;

// ---------------------------------------------------------------------------
// Zero-fill
// ---------------------------------------------------------------------------
__global__ void k_zero(float* __restrict__ p, size_t n) {
    size_t i = (size_t)blockIdx.x * blockDim.x + threadIdx.x;
    size_t stride = (size_t)gridDim.x * blockDim.x;
    for (; i < n; i += stride) p[i] = 0.0f;
}

// ---------------------------------------------------------------------------
// Degree accumulation: out_deg from src, in_deg from dst
// ---------------------------------------------------------------------------
__global__ void k_degrees(const int* __restrict__ src, const int* __restrict__ dst,
                          float* __restrict__ deg_s, float* __restrict__ deg_d, int E) {
    int e = blockIdx.x * blockDim.x + threadIdx.x;
    if (e >= E) return;
    atomicAdd(&deg_s[src[e]], 1.0f);
    atomicAdd(&deg_d[dst[e]], 1.0f);
}

// deg -> rsqrt(max(deg, 1))
__global__ void k_rsqrt(float* __restrict__ deg, size_t n) {
    size_t i = (size_t)blockIdx.x * blockDim.x + threadIdx.x;
    if (i >= n) return;
    deg[i] = rsqrtf(fmaxf(deg[i], 1.0f));
}

// ---------------------------------------------------------------------------
// Edge scatter: out[dst] += x[src] * (scale ? scale[src] : 1)
// 16 threads per edge, each thread handles 4 contiguous floats (float4).
// ---------------------------------------------------------------------------
__global__ void k_edge_scatter(const float* __restrict__ x, const float* __restrict__ scale,
                               const int* __restrict__ src, const int* __restrict__ dst,
                               float* __restrict__ out, int E) {
    int tid = blockIdx.x * blockDim.x + threadIdx.x;
    int e = tid >> 4;
    if (e >= E) return;
    int c = (tid & 15) << 2;
    int s = src[e], d = dst[e];
    float sc = scale ? scale[s] : 1.0f;
    const float4 v = *(const float4*)(x + (size_t)s * 64 + c);
    float* o = out + (size_t)d * 64 + c;
    atomicAdd(o + 0, v.x * sc);
    atomicAdd(o + 1, v.y * sc);
    atomicAdd(o + 2, v.z * sc);
    atomicAdd(o + 3, v.w * sc);
}

// ---------------------------------------------------------------------------
// Row-wise scale: h[i, :] *= a[i] * b[i]   (finishes layer0 right-norm and
// pre-applies layer1 left-norm in one pass). 16 threads/row, float4 each.
// ---------------------------------------------------------------------------
__global__ void k_row_scale2(float* __restrict__ h, const float* __restrict__ a,
                             const float* __restrict__ b, int N) {
    int tid = blockIdx.x * blockDim.x + threadIdx.x;
    int i = tid >> 4;
    if (i >= N) return;
    int c = (tid & 15) << 2;
    float s = a[i] * b[i];
    float4* p = (float4*)(h + (size_t)i * 64 + c);
    float4 v = *p;
    v.x *= s; v.y *= s; v.z *= s; v.w *= s;
    *p = v;
}

// ---------------------------------------------------------------------------
// Final edge message: out[dst] += h[src] * coef(e_feat)
// coef = 2 for even e_feat (0,2,4,6), else 1.
// ---------------------------------------------------------------------------
__global__ void k_final_scatter(const float* __restrict__ h, const int* __restrict__ src,
                                const int* __restrict__ dst, const int* __restrict__ ef,
                                float* __restrict__ out, int E) {
    int tid = blockIdx.x * blockDim.x + threadIdx.x;
    int e = tid >> 4;
    if (e >= E) return;
    int c = (tid & 15) << 2;
    int s = src[e], d = dst[e];
    float coef = (ef[e] & 1) ? 1.0f : 2.0f;
    const float4 v = *(const float4*)(h + (size_t)s * 64 + c);
    float* o = out + (size_t)d * 64 + c;
    atomicAdd(o + 0, v.x * coef);
    atomicAdd(o + 1, v.y * coef);
    atomicAdd(o + 2, v.z * coef);
    atomicAdd(o + 3, v.w * coef);
}

// ---------------------------------------------------------------------------
// WMMA GEMM: out[16 rows x 64 cols] = (rowscale[r] * A[r,:]) @ W + b
// One wave per 16-row tile. f32 WMMA 16x16x4, K = 64 -> 16 steps, 4 N-tiles.
//
// Fragment layouts per CDNA5 ISA (wave32):
//   A 16x4:  lanes 0-15 hold M=lane, lanes 16-31 hold M=lane-16;
//            VGPR0 = K {0 | 2}, VGPR1 = K {1 | 3} (split by lane half).
//   B 4x16:  lanes 0-15 hold N=lane (K 0/1), lanes 16-31 hold N=lane-16 (K 2/3).
//   C/D 16x16: VGPR j: lanes 0-15 -> M=j, lanes 16-31 -> M=j+8; N = lane&15.
// ---------------------------------------------------------------------------
__global__ void k_gemm_wmma(const float* __restrict__ A, const float* __restrict__ rowscale,
                            const float* __restrict__ W, const float* __restrict__ bias,
                            float* __restrict__ out, int N) {
    int wave = (blockIdx.x * blockDim.x + threadIdx.x) >> 5;
    int lane = threadIdx.x & 31;
    int rowBase = wave << 4;
    if (rowBase >= N) return;          // wave-uniform: EXEC stays all-ones below

    int mr = lane & 15;                // row-in-tile (A) / column-in-tile (B,D)
    int kh = lane >> 4;                // lane half -> K offset / M offset
    int row = rowBase + mr;
    int rowc = row < N ? row : (N - 1);
    float rs = rowscale[rowc];
    const float* Arow = A + (size_t)rowc * 64;

    // Bias-initialized accumulators: every acc VGPR's element shares column nt*16+mr.
    v8f acc[4];
#pragma unroll
    for (int nt = 0; nt < 4; ++nt) {
        float bb = bias[nt * 16 + mr];
#pragma unroll
        for (int j = 0; j < 8; ++j) acc[nt][j] = bb;
    }

#pragma unroll
    for (int kk = 0; kk < 16; ++kk) {
        int k0 = kk * 4 + kh * 2;      // this lane's K pair within the 4-wide step
        v2f a = *(const v2f*)(Arow + k0);   // 8B aligned (k0 even, row 256B aligned)
        a[0] *= rs;
        a[1] *= rs;
#pragma unroll
        for (int nt = 0; nt < 4; ++nt) {
            int c = nt * 16 + mr;
            v2f bf;
            bf[0] = W[(size_t)(k0 + 0) * 64 + c];
            bf[1] = W[(size_t)(k0 + 1) * 64 + c];
            acc[nt] = __builtin_amdgcn_wmma_f32_16x16x4_f32(
                /*neg_a=*/false, a, /*neg_b=*/false, bf,
                /*c_mod=*/(short)0, acc[nt], /*reuse_a=*/false, /*reuse_b=*/false);
        }
    }

    // Store D tiles: VGPR j -> row rowBase + j + 8*kh, column nt*16 + mr.
#pragma unroll
    for (int nt = 0; nt < 4; ++nt) {
#pragma unroll
        for (int j = 0; j < 8; ++j) {
            int r = rowBase + j + kh * 8;
            if (r < N) out[(size_t)r * 64 + nt * 16 + mr] = acc[nt][j];
        }
    }
}

// ---------------------------------------------------------------------------
// Launch
// ---------------------------------------------------------------------------
extern "C" void kernel_launch(void* const* d_in, const int* in_sizes, int n_in,
                              void* d_out, int out_size, void* d_ws, size_t ws_size,
                              hipStream_t stream) {
    const float* x   = (const float*)d_in[0];   // [N, 64]
    const float* W   = (const float*)d_in[1];   // [64, 64]
    const float* b   = (const float*)d_in[2];   // [64]
    const int*   src = (const int*)d_in[3];     // [E]
    const int*   dst = (const int*)d_in[4];     // [E]
    const int*   ef  = (const int*)d_in[5];     // [E]

    const int D = in_sizes[2];                  // 64
    const int N = in_sizes[0] / D;              // 50000
    const int E = in_sizes[3];                  // 800000
    float* out = (float*)d_out;

    // Workspace layout: [deg_s N][deg_d N][h0 N*D][h1 N*D]  (~26 MB for ref sizes)
    float* deg_s = (float*)d_ws;
    float* deg_d = deg_s + N;
    float* h0    = deg_d + N;
    float* h1    = h0 + (size_t)N * D;

    const size_t ws_floats = (size_t)2 * N + (size_t)2 * N * D;

    // 1) Zero scratch + output
    k_zero<<<2048, 256, 0, stream>>>(deg_s, ws_floats);
    k_zero<<<2048, 256, 0, stream>>>(out, (size_t)N * D);

    // 2) Degrees
    k_degrees<<<(E + 255) / 256, 256, 0, stream>>>(src, dst, deg_s, deg_d, E);

    // 3) deg -> rsqrt(max(deg,1)) over both contiguous arrays
    k_rsqrt<<<((size_t)2 * N + 255) / 256, 256, 0, stream>>>(deg_s, (size_t)2 * N);

    // 4) Layer-0 aggregation: h0[dst] += x[src] * dinv_src[src]
    k_edge_scatter<<<(E * 16 + 255) / 256, 256, 0, stream>>>(x, deg_s, src, dst, h0, E);

    // 5) h0[i,:] *= dinv_dst[i] * dinv_src[i]  (layer0 right-norm + layer1 left-norm)
    k_row_scale2<<<(N * 16 + 255) / 256, 256, 0, stream>>>(h0, deg_d, deg_s, N);

    // 6) Layer-1 aggregation: h1[dst] += h0[src]
    k_edge_scatter<<<(E * 16 + 255) / 256, 256, 0, stream>>>(h0, nullptr, src, dst, h1, E);

    // 7) h0 <- (dinv_dst[r] * h1[r,:]) @ W + b   (WMMA f32 16x16x4; h0 reused as output)
    {
        int waves = (N + 15) / 16;
        int blocks = (waves + 7) / 8;           // 8 waves (256 threads) per block
        k_gemm_wmma<<<blocks, 256, 0, stream>>>(h1, deg_d, W, b, h0, N);
    }

    // 8) Final message scatter: out[dst] += h0[src] * coef(e_feat)
    k_final_scatter<<<(E * 16 + 255) / 256, 256, 0, stream>>>(h0, src, dst, ef, out, E);
}